// DeformConv_66434554134598
// MI455X (gfx1250) — compile-verified
//
#include <hip/hip_runtime.h>
#include <hip/hip_bf16.h>

typedef __attribute__((ext_vector_type(2))) float v2f;
typedef __attribute__((ext_vector_type(8))) float v8f;

// ---- problem constants (from reference) ----
constexpr int BN   = 8;
constexpr int CIN  = 64;
constexpr int H    = 256;
constexpr int W    = 256;
constexpr int OC   = 128;
constexpr int KK   = 3;
constexpr int K2   = 9;
constexpr int STRIDE = 2;
constexpr int PAD  = 1;
constexpr int HO   = 128;
constexpr int WO   = 128;
constexpr int OFFC = 3 * K2;          // 27 offset-conv channels
constexpr int CK   = CIN * K2;        // 576 : GEMM K dimension
constexpr int NPIX = 32;              // pixels (N tile) per workgroup
constexpr int CKPAD = CK + 4;         // LDS row pad -> conflict-free b64 reads

// =====================================================================
// Kernel 1: offset conv  om[b][27][ho][wo] = conv3x3_s2(x, offset_w)+b
//           with sigmoid applied to mask channels 18..26
// =====================================================================
__global__ __launch_bounds__(256) void OffsetConvK(
    const float* __restrict__ x, const float* __restrict__ ow,
    const float* __restrict__ ob, float* __restrict__ om)
{
    __shared__ float wl[OFFC * CIN * K2];          // 62.2 KB of 320 KB LDS
    const int tid = threadIdx.x;
    for (int i = tid; i < OFFC * CIN * K2; i += 256) wl[i] = ow[i];
    __syncthreads();

    const int pid = blockIdx.x * 256 + tid;        // one output pixel / thread
    const int b  = pid >> 14;
    const int rm = pid & 16383;
    const int ho = rm >> 7;
    const int wo = rm & 127;

    float acc[OFFC];
#pragma unroll
    for (int o = 0; o < OFFC; ++o) acc[o] = ob[o];

    const int hbase = ho * STRIDE - PAD;
    const int wbase = wo * STRIDE - PAD;
    for (int ky = 0; ky < KK; ++ky) {
        const int hi = hbase + ky;
        if (hi < 0 || hi >= H) continue;
        for (int kx = 0; kx < KK; ++kx) {
            const int wi = wbase + kx;
            if (wi < 0 || wi >= W) continue;
            const float* xp = x + ((size_t)(b * CIN) * H + hi) * W + wi;
            const int wof = ky * KK + kx;
            for (int c = 0; c < CIN; ++c) {
                const float xv = xp[(size_t)c * H * W];
#pragma unroll
                for (int o = 0; o < OFFC; ++o)
                    acc[o] = fmaf(xv, wl[(o * CIN + c) * K2 + wof], acc[o]);
            }
        }
    }
#pragma unroll
    for (int o = 2 * K2; o < OFFC; ++o)            // mask channels -> sigmoid
        acc[o] = 1.0f / (1.0f + __expf(-acc[o]));

    float* op = om + ((size_t)(b * OFFC) * HO + ho) * WO + wo;
#pragma unroll
    for (int o = 0; o < OFFC; ++o) op[(size_t)o * HO * WO] = acc[o];
}

// =====================================================================
// Kernel 2: bilinear-sample im2col tile into LDS, then fp32 WMMA GEMM
//   out[b,oc,ho,wo] = sum_ck W[oc,ck] * col[ck,pix] + bias[oc]
//   per WG: 32 pixels x 128 oc, 8 waves. Each wave: 16 oc x 32 pixels
//   (two 16x16 D tiles sharing every A fragment) -> 2x A reuse,
//   2 x 144 accumulating v_wmma_f32_16x16x4_f32 along K=576.
// =====================================================================
__global__ __launch_bounds__(256) void DeformGemmK(
    const float* __restrict__ x, const float* __restrict__ om,
    const float* __restrict__ w2,      // (OC, CK) == weight reshaped
    const float* __restrict__ bias, float* __restrict__ out)
{
    __shared__ float colS[NPIX * CKPAD];           // 74.2 KB: [pixel][ck]
    __shared__ float offL[3 * K2 * NPIX];          // [kind][k2][pixel]

    const int tid      = threadIdx.x;
    const int pix_base = blockIdx.x * NPIX;
    const int wave = tid >> 5, lane = tid & 31;
    const int mn = lane & 15, half = lane >> 4;    // wave32 fragment coords
    const int oc_base = wave * 16;

    // pull this wave's weight rows toward L2/L0 early (global_prefetch_b8)
    __builtin_prefetch(w2 + (size_t)(oc_base + mn) * CK, 0, 3);

    // ---- Phase A: cache offsets/mask for these 32 pixels ----
    for (int i = tid; i < 3 * K2 * NPIX; i += 256) {
        const int ch = i / NPIX;                   // 0..26 = kind*9+k2
        const int p  = i % NPIX;
        const int pix = pix_base + p;
        const int b  = pix >> 14;
        const int ho = (pix >> 7) & 127;
        const int wo = pix & 127;
        offL[i] = om[((size_t)(b * OFFC + ch) * HO + ho) * WO + wo];
    }
    __syncthreads();

    // ---- Phase B: build masked bilinear im2col tile in LDS ----
    for (int e = tid; e < NPIX * CK; e += 256) {
        const int p  = e / CK;
        const int ck = e - p * CK;
        const int c  = ck / K2;
        const int k2 = ck - c * K2;
        const int pix = pix_base + p;
        const int b  = pix >> 14;
        const int ho = (pix >> 7) & 127;
        const int wo = pix & 127;

        const float off_x = offL[(0 * K2 + k2) * NPIX + p];
        const float off_y = offL[(1 * K2 + k2) * NPIX + p];
        const float msk   = offL[(2 * K2 + k2) * NPIX + p];

        const float py = off_y + (float)(k2 / 3) + (float)(ho * STRIDE - PAD);
        const float px = off_x + (float)(k2 % 3) + (float)(wo * STRIDE - PAD);
        const float fy0 = floorf(py), fx0 = floorf(px);
        const float wy1 = py - fy0, wx1 = px - fx0;
        const float wy0 = 1.0f - wy1, wx0 = 1.0f - wx1;
        const int iy0 = (int)fy0, ix0 = (int)fx0;

        const float* img = x + (size_t)(b * CIN + c) * H * W;
        float s = 0.0f;
        auto corner = [&](int yc, int xc, float wg) {
            if (yc >= 0 && yc < H && xc >= 0 && xc < W)
                s = fmaf(img[(size_t)yc * W + xc], wg, s);
        };
        corner(iy0,     ix0,     wy0 * wx0);
        corner(iy0,     ix0 + 1, wy0 * wx1);
        corner(iy0 + 1, ix0,     wy1 * wx0);
        corner(iy0 + 1, ix0 + 1, wy1 * wx1);

        colS[p * CKPAD + ck] = s * msk;
    }
    __syncthreads();

    // ---- Phase C: two 16x16 fp32 WMMA tiles per wave, K=576, step 4 ----
    // C/D layout: VGPR r <-> M = r + 8*half, N = mn
    v8f acc0, acc1;
#pragma unroll
    for (int r = 0; r < 8; ++r) {
        const float bv = bias[oc_base + r + half * 8];
        acc0[r] = bv;
        acc1[r] = bv;
    }

    // A layout: m = mn, holds K = k0 + 2*half .. +1  (contiguous b64)
    const float* arow  = w2 + (size_t)(oc_base + mn) * CK + half * 2;
    // B layout: n = mn, holds K = k0 + 2*half .. +1  (contiguous padded rows)
    const float* brow0 = &colS[(mn     ) * CKPAD + half * 2];   // pixels 0..15
    const float* brow1 = &colS[(mn + 16) * CKPAD + half * 2];   // pixels 16..31

#pragma unroll 4
    for (int k0 = 0; k0 < CK; k0 += 4) {
        v2f a   = *(const v2f*)(arow  + k0);
        v2f bf0 = *(const v2f*)(brow0 + k0);
        v2f bf1 = *(const v2f*)(brow1 + k0);
        acc0 = __builtin_amdgcn_wmma_f32_16x16x4_f32(
            false, a, false, bf0, (short)0, acc0, false, false);
        acc1 = __builtin_amdgcn_wmma_f32_16x16x4_f32(
            false, a, false, bf1, (short)0, acc1, false, false);
    }

    // ---- store both D tiles ----
    {
        const int pix = pix_base + mn;             // pixels 0..15 group
        const int b  = pix >> 14;
        const int ho = (pix >> 7) & 127;
        const int wo = pix & 127;
#pragma unroll
        for (int r = 0; r < 8; ++r) {
            const int oc = oc_base + r + half * 8;
            out[((size_t)(b * OC + oc) * HO + ho) * WO + wo] = acc0[r];
        }
    }
    {
        const int pix = pix_base + mn + 16;        // pixels 16..31 group
        const int b  = pix >> 14;
        const int ho = (pix >> 7) & 127;
        const int wo = pix & 127;
#pragma unroll
        for (int r = 0; r < 8; ++r) {
            const int oc = oc_base + r + half * 8;
            out[((size_t)(b * OC + oc) * HO + ho) * WO + wo] = acc1[r];
        }
    }
}

// =====================================================================
extern "C" void kernel_launch(void* const* d_in, const int* in_sizes, int n_in,
                              void* d_out, int out_size, void* d_ws, size_t ws_size,
                              hipStream_t stream) {
    const float* x        = (const float*)d_in[0];   // (8,64,256,256)
    const float* offset_w = (const float*)d_in[1];   // (27,64,3,3)
    const float* offset_b = (const float*)d_in[2];   // (27,)
    const float* weight   = (const float*)d_in[3];   // (128,64,3,3) == (128,576)
    const float* bias     = (const float*)d_in[4];   // (128,)
    float*       out      = (float*)d_out;           // (8,128,128,128)

    float* om = (float*)d_ws;                        // (8,27,128,128) = 14.2 MB

    const int npix_total = BN * HO * WO;             // 131072
    OffsetConvK<<<npix_total / 256, 256, 0, stream>>>(x, offset_w, offset_b, om);
    DeformGemmK<<<npix_total / NPIX, 256, 0, stream>>>(x, om, weight, bias, out);
}